// Net_23029614641786
// MI455X (gfx1250) — compile-verified
//
#include <hip/hip_runtime.h>
#include <hip/hip_bf16.h>

typedef __attribute__((ext_vector_type(2))) float v2f;
typedef __attribute__((ext_vector_type(8))) float v8f;

#define T_STEPS 255
#define BATCH   256
#define KDIM    784      // 196 k-blocks of 4
#define NOUT    40
#define NPAD    48       // 3 n-tiles of 16
#define KPAIRS  (KDIM/2) // 392

// ---------------------------------------------------------------------------
// Kernel 1: CUR = X @ W^T over all timesteps at once.
//   X:   [T*B, 784] f32 (row-major)
//   W:   [40, 784]  f32 (row-major)
//   CUR: [T*B, 40]  f32 (workspace)
// One workgroup (4 waves / 128 threads) per timestep; each wave computes a
// 64-row x 48-col tile via V_WMMA_F32_16X16X4_F32 (fp32 matrix pipe).
// ---------------------------------------------------------------------------
__global__ __launch_bounds__(128) void snn_gemm_kernel(
    const float* __restrict__ x, const float* __restrict__ W,
    float* __restrict__ cur)
{
    // W staged in LDS, k-pair interleaved and padded to 48 output columns:
    // Wp[kp][n] = { W[n][2*kp], W[n][2*kp+1] }   (zeros for n >= 40)
    __shared__ v2f Wp[KPAIRS][NPAD];  // 392*48*8 = 150528 B (< 320 KB WGP LDS)

    const int tid = threadIdx.x;
    const int t   = blockIdx.x;

    const v2f* Wv = (const v2f*)W;    // Wv[n*392 + kp]
    for (int i = tid; i < KPAIRS * NPAD; i += 128) {  // 18816 = 147*128, no divergence
        const int kp = i / NPAD;
        const int n  = i % NPAD;
        v2f v;
        if (n < NOUT) v = Wv[n * KPAIRS + kp];
        else          { v.x = 0.0f; v.y = 0.0f; }
        Wp[kp][n] = v;
    }
    __syncthreads();

    const int wave = tid >> 5;
    const int lane = tid & 31;
    const int h    = lane >> 4;   // half-wave (selects K sub-pair)
    const int r    = lane & 15;   // row/col within tile

    const int row_base = t * BATCH + wave * 64;   // 4 m-tiles of 16 rows

    // Per-lane A pointers: lane (h, r) of m-tile mt streams row (base+mt*16+r),
    // columns {4kb+2h, 4kb+2h+1} -> one contiguous 8-byte load per k-block.
    const float* ap[4];
#pragma unroll
    for (int mt = 0; mt < 4; ++mt)
        ap[mt] = x + (size_t)(row_base + mt * 16 + r) * KDIM + 2 * h;

    v8f acc[4][3] = {};

    for (int kb = 0; kb < KDIM / 4; ++kb) {
        const int kp = 2 * kb + h;
        const v2f b0 = Wp[kp][r];          // conflict-free ds_load_b64
        const v2f b1 = Wp[kp][16 + r];
        const v2f b2 = Wp[kp][32 + r];
#pragma unroll
        for (int mt = 0; mt < 4; ++mt) {
            const v2f a = *(const v2f*)(ap[mt] + 4 * kb);
            acc[mt][0] = __builtin_amdgcn_wmma_f32_16x16x4_f32(
                false, a, false, b0, (short)0, acc[mt][0], false, false);
            acc[mt][1] = __builtin_amdgcn_wmma_f32_16x16x4_f32(
                false, a, false, b1, (short)0, acc[mt][1], false, false);
            acc[mt][2] = __builtin_amdgcn_wmma_f32_16x16x4_f32(
                false, a, false, b2, (short)0, acc[mt][2], false, false);
        }
    }

    // C/D layout: VGPR g of lane (h, r) holds C[M = 8h+g][N = r] of the tile.
#pragma unroll
    for (int mt = 0; mt < 4; ++mt) {
#pragma unroll
        for (int nt = 0; nt < 3; ++nt) {
            const int n = nt * 16 + r;
            if (n < NOUT) {
#pragma unroll
                for (int g = 0; g < 8; ++g) {
                    const int row = row_base + mt * 16 + h * 8 + g;
                    cur[(size_t)row * NOUT + n] = acc[mt][nt][g];
                }
            }
        }
    }
}

// ---------------------------------------------------------------------------
// Kernel 2: LIF scan over time. One thread per (batch, neuron) element.
//   mem_new = (0.95*mem + cur) * (1 - (mem > 1))   (reset from PREVIOUS mem)
//   spk     = (mem_new - 1 > 0)
// Outputs concatenated: [T,B,40] spikes then [T,B,40] membrane.
// ---------------------------------------------------------------------------
__global__ __launch_bounds__(256) void snn_scan_kernel(
    const float* __restrict__ cur, float* __restrict__ out)
{
    const int idx    = blockIdx.x * blockDim.x + threadIdx.x;   // 0..10239
    const int stride = BATCH * NOUT;                            // 10240
    float* __restrict__ spk_out = out;
    float* __restrict__ mem_out = out + (size_t)T_STEPS * stride;

    float mem = 0.0f;
    for (int t = 0; t < T_STEPS; ++t) {
        const float c     = cur[(size_t)t * stride + idx];
        const float reset = (mem > 1.0f) ? 1.0f : 0.0f;
        mem = (0.95f * mem + c) * (1.0f - reset);
        spk_out[(size_t)t * stride + idx] = (mem > 1.0f) ? 1.0f : 0.0f;
        mem_out[(size_t)t * stride + idx] = mem;
    }
}

extern "C" void kernel_launch(void* const* d_in, const int* in_sizes, int n_in,
                              void* d_out, int out_size, void* d_ws, size_t ws_size,
                              hipStream_t stream)
{
    (void)in_sizes; (void)n_in; (void)out_size; (void)ws_size;
    const float* x = (const float*)d_in[0];   // [255, 256, 784] f32
    const float* W = (const float*)d_in[1];   // [40, 784] f32
    float* out = (float*)d_out;               // [2, 255, 256, 40] f32 (spk, mem)
    float* cur = (float*)d_ws;                // [255*256, 40] f32 scratch (10.4 MB)

    snn_gemm_kernel<<<dim3(T_STEPS), dim3(128), 0, stream>>>(x, W, cur);
    snn_scan_kernel<<<dim3((BATCH * NOUT) / 256), dim3(256), 0, stream>>>(cur, out);
}